// Dot_Product_Attention_49263274885448
// MI455X (gfx1250) — compile-verified
//
#include <hip/hip_runtime.h>
#include <math.h>

typedef __attribute__((ext_vector_type(2))) float v2f;
typedef __attribute__((ext_vector_type(8))) float v8f;

#define BATCH 64
#define SEQ   4096
#define HD    256
#define CHUNK 512
#define NCHUNK (SEQ / CHUNK)

// ---------------------------------------------------------------------------
// Small fp32 GEMM via V_WMMA_F32_16X16X4_F32.
// C[64x256] = A[64x256] * B[256x256] * scale, B optionally transposed
// (transB: B[k,n] = Bm[n*256+k], else Bm[k*256+n]).
// grid = 64 blocks (4 M-tiles x 16 N-tiles), 32 threads (1 wave) each.
// A-frag 16x4: lane<16 -> M=lane, {v0=K0,v1=K1}; lane>=16 -> {K2,K3}.
// B-frag 4x16: lane<16 -> N=lane, {v0=K0,v1=K1}; lane>=16 -> {K2,K3}.
// C/D 16x16:  vgpr r, lanes 0-15 -> (M=r, N=lane); lanes 16-31 -> (M=r+8).
// ---------------------------------------------------------------------------
__global__ void gemm64_wmma(const float* __restrict__ A,
                            const float* __restrict__ Bm,
                            float* __restrict__ C,
                            int transB, float scale) {
    const int lane = threadIdx.x & 31;
    const int m0   = (blockIdx.x >> 4) << 4;
    const int n0   = (blockIdx.x & 15) << 4;
    const int half = lane >> 4;        // 0 or 1
    const int l    = lane & 15;
    v8f acc = {};
    for (int k0 = 0; k0 < HD; k0 += 4) {
        const int ka = k0 + half * 2;
        v2f a, b;
        a.x = A[(m0 + l) * HD + ka];
        a.y = A[(m0 + l) * HD + ka + 1];
        const int col = n0 + l;
        if (transB) {
            b.x = Bm[col * HD + ka];
            b.y = Bm[col * HD + ka + 1];
        } else {
            b.x = Bm[ka * HD + col];
            b.y = Bm[(ka + 1) * HD + col];
        }
        acc = __builtin_amdgcn_wmma_f32_16x16x4_f32(
            /*neg_a=*/false, a, /*neg_b=*/false, b,
            /*c_mod=*/(short)0, acc, /*reuse_a=*/false, /*reuse_b=*/false);
    }
#pragma unroll
    for (int r = 0; r < 8; ++r) {
        const int row = m0 + r + half * 8;
        C[row * HD + (n0 + l)] = acc[r] * scale;
    }
}

// ---------------------------------------------------------------------------
// Streaming kernel: per (chunk, batch) block of 256 threads (8 waves):
//   scores[s] = t[b]·key[b,s]  (t pre-scaled by 1/sqrt(H))
//   mask==true -> -1e15 ; chunk-local softmax numerator written to p_out,
//   chunk stats (max, sumexp) and partial u = sum_s e_s * value[b,s,:] to ws.
// Key/value loads are b128 per lane; all global traffic is coalesced.
// ---------------------------------------------------------------------------
__global__ void attn_chunk(const float* __restrict__ key,
                           const float* __restrict__ value,
                           const unsigned char* __restrict__ mask,  // jnp bool = 1 byte
                           const float* __restrict__ T,             // 64x256
                           float* __restrict__ p_out,               // 64x4096 (unnormalized)
                           float* __restrict__ u_chunks,            // 64 x NCHUNK x 256
                           float* __restrict__ stats) {             // 64 x NCHUNK x 2
    __shared__ float  lds_t[HD];
    __shared__ float  lds_sc[CHUNK];
    __shared__ float  redbuf[256];
    __shared__ float4 lds_u[256];

    const int b    = blockIdx.y;
    const int c    = blockIdx.x;
    const int tid  = threadIdx.x;
    const int lane = tid & 31;
    const int wave = tid >> 5;
    const int s0   = c * CHUNK;

    lds_t[tid] = T[b * HD + tid];
    // mask bytes for this chunk, coalesced (2 per thread)
    const unsigned char mk0 = mask[(size_t)b * SEQ + s0 + tid];
    const unsigned char mk1 = mask[(size_t)b * SEQ + s0 + tid + 256];
    __syncthreads();

    // ---- scores: each wave handles 64 consecutive s values ----
    const float*  keyb = key + (size_t)b * SEQ * HD;
    const float4* tp   = (const float4*)lds_t;
    const float4  t0v  = tp[lane];
    const float4  t1v  = tp[lane + 32];
    for (int i = 0; i < 64; ++i) {
        const int s = s0 + wave * 64 + i;
        const float4* kp = (const float4*)(keyb + (size_t)s * HD);
        const float4 k0 = kp[lane];        // d = lane*4 .. +3
        const float4 k1 = kp[lane + 32];   // d = 128 + lane*4 ..
        float d = k0.x * t0v.x + k0.y * t0v.y + k0.z * t0v.z + k0.w * t0v.w
                + k1.x * t1v.x + k1.y * t1v.y + k1.z * t1v.z + k1.w * t1v.w;
#pragma unroll
        for (int off = 16; off > 0; off >>= 1) d += __shfl_xor(d, off, 32);
        if (lane == 0) lds_sc[wave * 64 + i] = d;   // raw score (T carries 1/sqrt(H))
    }
    __syncthreads();

    // ---- apply mask + chunk max ----
    const float sc0 = mk0 ? -1.0e15f : lds_sc[tid];
    const float sc1 = mk1 ? -1.0e15f : lds_sc[tid + 256];
    redbuf[tid] = fmaxf(sc0, sc1);
    __syncthreads();
    for (int st = 128; st > 0; st >>= 1) {
        if (tid < st) redbuf[tid] = fmaxf(redbuf[tid], redbuf[tid + st]);
        __syncthreads();
    }
    const float mc = redbuf[0];
    __syncthreads();

    // ---- exp + chunk sum + unnormalized p ----
    const float e0 = __expf(sc0 - mc);
    const float e1 = __expf(sc1 - mc);
    lds_sc[tid]       = e0;
    lds_sc[tid + 256] = e1;
    p_out[(size_t)b * SEQ + s0 + tid]       = e0;
    p_out[(size_t)b * SEQ + s0 + tid + 256] = e1;
    redbuf[tid] = e0 + e1;
    __syncthreads();
    for (int st = 128; st > 0; st >>= 1) {
        if (tid < st) redbuf[tid] += redbuf[tid + st];
        __syncthreads();
    }
    if (tid == 0) {
        stats[((size_t)b * NCHUNK + c) * 2 + 0] = mc;
        stats[((size_t)b * NCHUNK + c) * 2 + 1] = redbuf[0];
    }

    // ---- partial u, b128 per lane ----
    // thread = (q, d4): q in 0..3 partitions s; d4 in 0..63 owns dims 4*d4..4*d4+3
    const int d4 = tid & 63;
    const int q  = tid >> 6;
    const float4* valb4 = (const float4*)(value + (size_t)b * SEQ * HD) + d4;
    float4 acc = make_float4(0.f, 0.f, 0.f, 0.f);
#pragma unroll 4
    for (int j = q; j < CHUNK; j += 4) {
        const float4 v4 = valb4[(size_t)(s0 + j) * (HD / 4)];
        const float  e  = lds_sc[j];
        acc.x = fmaf(e, v4.x, acc.x);
        acc.y = fmaf(e, v4.y, acc.y);
        acc.z = fmaf(e, v4.z, acc.z);
        acc.w = fmaf(e, v4.w, acc.w);
    }
    lds_u[tid] = acc;
    __syncthreads();
    if (tid < 64) {
        const float4 a0 = lds_u[tid];
        const float4 a1 = lds_u[tid + 64];
        const float4 a2 = lds_u[tid + 128];
        const float4 a3 = lds_u[tid + 192];
        float4 r;
        r.x = a0.x + a1.x + a2.x + a3.x;
        r.y = a0.y + a1.y + a2.y + a3.y;
        r.z = a0.z + a1.z + a2.z + a3.z;
        r.w = a0.w + a1.w + a2.w + a3.w;
        ((float4*)(u_chunks + ((size_t)b * NCHUNK + c) * HD))[tid] = r;
    }
}

// ---------------------------------------------------------------------------
// Combine chunk stats: global max/sum, rescale p_attn, build final u/Z.
// grid = 64 blocks, 256 threads.
// ---------------------------------------------------------------------------
__global__ void attn_combine(const float* __restrict__ stats,
                             const float* __restrict__ u_chunks,
                             float* __restrict__ p_out,
                             float* __restrict__ U) {
    const int b   = blockIdx.x;
    const int tid = threadIdx.x;
    __shared__ float sw[NCHUNK];
    __shared__ float Zb;
    if (tid == 0) {
        float M = -INFINITY;
        float m[NCHUNK], z[NCHUNK];
        for (int c = 0; c < NCHUNK; ++c) {
            m[c] = stats[((size_t)b * NCHUNK + c) * 2 + 0];
            z[c] = stats[((size_t)b * NCHUNK + c) * 2 + 1];
            M = fmaxf(M, m[c]);
        }
        float Z = 0.0f;
        for (int c = 0; c < NCHUNK; ++c) {
            sw[c] = __expf(m[c] - M);
            Z += sw[c] * z[c];
        }
        Zb = Z;
    }
    __syncthreads();
    const float invZ = 1.0f / Zb;

    float acc = 0.0f;
    for (int c = 0; c < NCHUNK; ++c)
        acc += sw[c] * u_chunks[((size_t)b * NCHUNK + c) * HD + tid];
    U[(size_t)b * HD + tid] = acc * invZ;

    for (int s = tid; s < SEQ; s += 256) {
        const int c = s >> 9;   // s / CHUNK
        p_out[(size_t)b * SEQ + s] *= sw[c] * invZ;
    }
}

// ---------------------------------------------------------------------------
extern "C" void kernel_launch(void* const* d_in, const int* in_sizes, int n_in,
                              void* d_out, int out_size, void* d_ws, size_t ws_size,
                              hipStream_t stream) {
    const float*         query = (const float*)d_in[0];         // 64 x 1 x 256
    const float*         key   = (const float*)d_in[1];         // 64 x 4096 x 256
    const float*         value = (const float*)d_in[2];         // 64 x 4096 x 256
    const unsigned char* mask  = (const unsigned char*)d_in[3]; // 64 x 4096 (bool)
    const float*         WQ    = (const float*)d_in[4];         // 256 x 256
    const float*         WK    = (const float*)d_in[5];
    const float*         WV    = (const float*)d_in[6];

    float* out_o = (float*)d_out;                 // 64 x 256
    float* p_o   = (float*)d_out + BATCH * HD;    // 64 x 4096

    float* ws = (float*)d_ws;
    float* Q        = ws;                                  // 16384
    float* T        = Q + BATCH * HD;                      // 16384
    float* u_chunks = T + BATCH * HD;                      // 64*8*256 = 131072
    float* stats    = u_chunks + BATCH * NCHUNK * HD;      // 1024
    float* U        = stats + BATCH * NCHUNK * 2;          // 16384

    // Q = query * WQ^T
    gemm64_wmma<<<dim3(64), dim3(32), 0, stream>>>(query, WQ, Q, /*transB=*/1, 1.0f);
    // T = Q * WK, pre-scaled by 1/sqrt(H) = 1/16
    gemm64_wmma<<<dim3(64), dim3(32), 0, stream>>>(Q, WK, T, /*transB=*/0, 0.0625f);
    // scores + chunked softmax + partial value reduction (streams key & value once)
    attn_chunk<<<dim3(NCHUNK, BATCH), dim3(256), 0, stream>>>(
        key, value, mask, T, p_o, u_chunks, stats);
    // merge chunks, normalize p_attn, build U
    attn_combine<<<dim3(BATCH), dim3(256), 0, stream>>>(stats, u_chunks, p_o, U);
    // out = U * WV^T
    gemm64_wmma<<<dim3(64), dim3(32), 0, stream>>>(U, WV, out_o, /*transB=*/1, 1.0f);
}